// AtomAtomAffinities_17051020165714
// MI455X (gfx1250) — compile-verified
//
#include <hip/hip_runtime.h>

// ---------------------------------------------------------------------------
// AtomAtomAffinities — CDNA5 (gfx1250) implementation.
// Memory-bound (~1.7 GB @ 23.3 TB/s ≈ 73 us; only ~90 GFLOP fp32), so all
// GEMMs use full-precision V_WMMA_F32_16X16X4_F32 (exact fp32 accumulate).
// FC head folded into edge projection so the [E,128] intermediate stays in
// registers; segment-sum done in LDS bins (G=4096 fits in 16 KB).
// ---------------------------------------------------------------------------

#define F_IN   256   // node feature dim
#define FE_IN  64    // edge feature dim
#define H_DIM  128   // hidden dim
#define G_MAX  4096  // graph bins held in LDS
#define WPAD   4     // LDS row padding (floats) -> conflict-free B-frag reads
#define EPAD   4

typedef float v2f __attribute__((ext_vector_type(2)));
typedef float v8f __attribute__((ext_vector_type(8)));

static inline int imin(int a, int b) { return a < b ? a : b; }

// --------------------------------------------------------------------------
__global__ void zero_out_kernel(float* __restrict__ p, int n) {
    int i = blockIdx.x * blockDim.x + threadIdx.x;
    if (i < n) p[i] = 0.0f;
}

// We'[h][k] = Wf[h] * We[h][k];  be'[h] = Wf[h] * be[h]
__global__ void fold_w_kernel(const float* __restrict__ We, const float* __restrict__ be,
                              const float* __restrict__ Wf,
                              float* __restrict__ Wo, float* __restrict__ bo) {
    int i = blockIdx.x * blockDim.x + threadIdx.x;
    if (i < H_DIM * FE_IN) {
        int h = i / FE_IN;
        Wo[i] = Wf[h] * We[i];
    }
    if (i < H_DIM) bo[i] = Wf[i] * be[i];
}

// --------------------------------------------------------------------------
// Y[N,128] = X[N,256] @ W[128,256]^T + b, fp32 WMMA 16x16x4.
// One workgroup = 8 waves; wave w owns N-block n0 = 16*w. Full W + bias
// cached in (dynamic) LDS once per workgroup; grid-stride over 16-row tiles.
// LDS: 128*(256+4) + 128 + 16*(256+4)  floats  ~= 147 KB (< 160 KB/WG).
// --------------------------------------------------------------------------
__global__ void proj_gemm_kernel(const float* __restrict__ X, int N,
                                 const float* __restrict__ W, const float* __restrict__ b,
                                 float* __restrict__ Y, int nTiles) {
    extern __shared__ float lds[];
    float* Ws = lds;                              // [128][256+WPAD]
    float* bs = Ws + H_DIM * (F_IN + WPAD);       // [128]
    float* Xs = bs + H_DIM;                       // [16][256+WPAD]

    const int tid = threadIdx.x;

    // Stage W (row-padded) + bias.
    for (int i4 = tid; i4 < H_DIM * F_IN / 4; i4 += blockDim.x) {
        int row = i4 / (F_IN / 4);
        int c4  = i4 % (F_IN / 4);
        float4 v = ((const float4*)W)[i4];
        *((float4*)(Ws + row * (F_IN + WPAD) + c4 * 4)) = v;
    }
    if (tid < H_DIM) bs[tid] = b[tid];
    __syncthreads();

    const int lane = tid & 31;
    const int wave = tid >> 5;        // 0..7  -> N block
    const int hf   = lane >> 4;       // half-wave: 0/1
    const int ln   = lane & 15;
    const int n0   = wave * 16;
    const int koff = hf * 2;          // ISA 32-bit A/B fragment K offset
    const float bn = bs[n0 + ln];

    for (int t = blockIdx.x; t < nTiles; t += gridDim.x) {
        const int row0 = t * 16;
        __syncthreads();   // previous tile's LDS reads done
        // Stage 16x256 A tile.
        for (int i4 = tid; i4 < 16 * F_IN / 4; i4 += blockDim.x) {
            int r  = i4 / (F_IN / 4);
            int c4 = i4 % (F_IN / 4);
            int gr = row0 + r; if (gr >= N) gr = N - 1;
            float4 v = ((const float4*)(X + (size_t)gr * F_IN))[c4];
            *((float4*)(Xs + r * (F_IN + WPAD) + c4 * 4)) = v;
        }
        __syncthreads();

        v8f c = {0.f, 0.f, 0.f, 0.f, 0.f, 0.f, 0.f, 0.f};
        #pragma unroll
        for (int k = 0; k < F_IN; k += 4) {
            v2f a, bb;
            a.x  = Xs[ln * (F_IN + WPAD) + k + koff];
            a.y  = Xs[ln * (F_IN + WPAD) + k + koff + 1];
            bb.x = Ws[(n0 + ln) * (F_IN + WPAD) + k + koff];
            bb.y = Ws[(n0 + ln) * (F_IN + WPAD) + k + koff + 1];
            c = __builtin_amdgcn_wmma_f32_16x16x4_f32(false, a, false, bb,
                                                      (short)0, c, false, false);
        }
        // D layout: VGPR r -> row (r + 8*hf), col n0+ln. Coalesced 64B stores.
        #pragma unroll
        for (int r = 0; r < 8; ++r) {
            int m = row0 + r + hf * 8;
            if (m < N) Y[(size_t)m * H_DIM + n0 + ln] = c[r] + bn;
        }
    }
}

// --------------------------------------------------------------------------
// Fused edge pass: e' = EF @ We'^T + be' (WMMA, in registers),
// logit_e = dot(e', HS[src] * HD[dst]) + bf, segment-sum into LDS bins,
// one global atomic per nonzero bin at the end.
// 8 waves/WG; each wave owns 16 edges per iteration.
// LDS: We' 34.8KB + bins 16KB + 8 edge slabs 34.8KB + bias ~= 86 KB.
// --------------------------------------------------------------------------
__global__ void edge_fused_kernel(const float* __restrict__ EF,
                                  const int* __restrict__ srcI,
                                  const int* __restrict__ dstI,
                                  const int* __restrict__ gidI,
                                  const float* __restrict__ Wef,
                                  const float* __restrict__ bef,
                                  const float* __restrict__ HS,
                                  const float* __restrict__ HD,
                                  const float* __restrict__ bfp,
                                  float* __restrict__ OUT, int E, int G) {
    extern __shared__ float lds[];
    float* Wes  = lds;                                  // [128][64+EPAD]
    float* bes  = Wes + H_DIM * (FE_IN + EPAD);         // [128]
    float* accs = bes + H_DIM;                          // [G_MAX]
    float* efs  = accs + G_MAX;                         // 8 * [16][64+EPAD]

    const int tid = threadIdx.x;
    for (int g = tid; g < G_MAX; g += blockDim.x) accs[g] = 0.0f;
    for (int i4 = tid; i4 < H_DIM * FE_IN / 4; i4 += blockDim.x) {
        int row = i4 / (FE_IN / 4);
        int c4  = i4 % (FE_IN / 4);
        float4 v = ((const float4*)Wef)[i4];
        *((float4*)(Wes + row * (FE_IN + EPAD) + c4 * 4)) = v;
    }
    if (tid < H_DIM) bes[tid] = bef[tid];
    __syncthreads();

    const int lane = tid & 31;
    const int wave = tid >> 5;
    const int hf   = lane >> 4;
    const int ln   = lane & 15;
    const int koff = hf * 2;
    const float bfv = bfp[0];
    float* myefs = efs + wave * 16 * (FE_IN + EPAD);

    float besv[8];
    #pragma unroll
    for (int nb = 0; nb < 8; ++nb) besv[nb] = bes[nb * 16 + ln];

    const int nTiles = (E + 15) >> 4;
    const int stride = gridDim.x * 8;

    for (int t = blockIdx.x * 8 + wave; t < nTiles; t += stride) {
        const int e0 = t * 16;
        // Stage this wave's 16x64 edge-feature tile (private slab -> no barrier).
        {
            int r  = lane >> 1;
            int hc = lane & 1;
            int ge = e0 + r; if (ge >= E) ge = E - 1;
            const float4* sp = (const float4*)(EF + (size_t)ge * FE_IN + hc * 32);
            float4* dp = (float4*)(myefs + r * (FE_IN + EPAD) + hc * 32);
            #pragma unroll
            for (int q = 0; q < 8; ++q) dp[q] = sp[q];
        }
        // Edge metadata: lane (ln) holds edge e0+ln in lanes 0..15 (dup in 16..31).
        int sv, dv, gv;
        {
            int ge = e0 + ln; if (ge >= E) ge = E - 1;
            sv = srcI[ge]; dv = dstI[ge]; gv = gidI[ge];
        }

        v8f c[8];
        #pragma unroll
        for (int nb = 0; nb < 8; ++nb) c[nb] = 0.0f;

        #pragma unroll
        for (int k = 0; k < FE_IN; k += 4) {
            v2f a;
            a.x = myefs[ln * (FE_IN + EPAD) + k + koff];
            a.y = myefs[ln * (FE_IN + EPAD) + k + koff + 1];
            #pragma unroll
            for (int nb = 0; nb < 8; ++nb) {
                v2f bb;
                bb.x = Wes[(nb * 16 + ln) * (FE_IN + EPAD) + k + koff];
                bb.y = Wes[(nb * 16 + ln) * (FE_IN + EPAD) + k + koff + 1];
                c[nb] = __builtin_amdgcn_wmma_f32_16x16x4_f32(false, a, false, bb,
                                                              (short)0, c[nb], false, false);
            }
        }

        // Per edge r: this lane holds column h = nb*16+ln of e'; gather the
        // matching hs/hd values (16 consecutive floats -> 64B coalesced, L2-hit),
        // reduce over the half-wave, bin into LDS.
        #pragma unroll
        for (int r = 0; r < 8; ++r) {
            const int rsel = r + hf * 8;             // edge index within tile
            int s_e = __shfl(sv, rsel);
            int d_e = __shfl(dv, rsel);
            int g_e = __shfl(gv, rsel);
            float part = 0.0f;
            #pragma unroll
            for (int nb = 0; nb < 8; ++nb) {
                const int h = nb * 16 + ln;
                float eh = c[nb][r] + besv[nb];
                part += eh * HS[(size_t)s_e * H_DIM + h] * HD[(size_t)d_e * H_DIM + h];
            }
            part += __shfl_xor(part, 1);
            part += __shfl_xor(part, 2);
            part += __shfl_xor(part, 4);
            part += __shfl_xor(part, 8);             // half-wave sum (width 16)
            if (ln == 0 && (e0 + rsel) < E)
                atomicAdd(&accs[g_e], part + bfv);   // ds_add_f32
        }
    }

    __syncthreads();
    for (int g = tid; g < G && g < G_MAX; g += blockDim.x) {
        float v = accs[g];
        if (v != 0.0f) atomicAdd(&OUT[g], v);
    }
}

// --------------------------------------------------------------------------
extern "C" void kernel_launch(void* const* d_in, const int* in_sizes, int n_in,
                              void* d_out, int out_size, void* d_ws, size_t ws_size,
                              hipStream_t stream) {
    const float* ligand_h = (const float*)d_in[0];
    const float* pocket_h = (const float*)d_in[1];
    const float* e_lp     = (const float*)d_in[2];
    const float* e_pl     = (const float*)d_in[3];
    const int*   src_lp   = (const int*)d_in[4];
    const int*   dst_lp   = (const int*)d_in[5];
    const int*   src_pl   = (const int*)d_in[6];
    const int*   dst_pl   = (const int*)d_in[7];
    const int*   gid_lp   = (const int*)d_in[8];
    const int*   gid_pl   = (const int*)d_in[9];
    // d_in[10] = num_graphs scalar (device); G derived from out_size instead.
    const float* W_lp_src = (const float*)d_in[11];
    const float* b_lp_src = (const float*)d_in[12];
    const float* W_lp_dst = (const float*)d_in[13];
    const float* b_lp_dst = (const float*)d_in[14];
    const float* W_lp_e   = (const float*)d_in[15];
    const float* b_lp_e   = (const float*)d_in[16];
    const float* W_pl_src = (const float*)d_in[17];
    const float* b_pl_src = (const float*)d_in[18];
    const float* W_pl_dst = (const float*)d_in[19];
    const float* b_pl_dst = (const float*)d_in[20];
    const float* W_pl_e   = (const float*)d_in[21];
    const float* b_pl_e   = (const float*)d_in[22];
    const float* W_fc_lp  = (const float*)d_in[23];
    const float* b_fc_lp  = (const float*)d_in[24];
    const float* W_fc_pl  = (const float*)d_in[25];
    const float* b_fc_pl  = (const float*)d_in[26];

    const int NL  = in_sizes[0] / F_IN;
    const int NP  = in_sizes[1] / F_IN;
    const int ELP = in_sizes[4];
    const int EPL = in_sizes[6];
    const int G   = out_size / 2;

    // Workspace layout (floats): 4 projected node arrays (~307 MB) + folded W.
    float* ws      = (float*)d_ws;
    float* h_lp_s  = ws;                                  // [NL,128] ligand @ W_lp_src
    float* h_lp_d  = h_lp_s + (size_t)NL * H_DIM;         // [NP,128] pocket @ W_lp_dst
    float* h_pl_s  = h_lp_d + (size_t)NP * H_DIM;         // [NP,128] pocket @ W_pl_src
    float* h_pl_d  = h_pl_s + (size_t)NP * H_DIM;         // [NL,128] ligand @ W_pl_dst
    float* we_lp   = h_pl_d + (size_t)NL * H_DIM;         // [128,64]
    float* be_lp   = we_lp + H_DIM * FE_IN;               // [128]
    float* we_pl   = be_lp + H_DIM;                       // [128,64]
    float* be_pl   = we_pl + H_DIM * FE_IN;               // [128]
    (void)ws_size; (void)n_in;

    // Graph replays accumulate into d_out -> must zero every call.
    zero_out_kernel<<<(2 * G + 255) / 256, 256, 0, stream>>>((float*)d_out, 2 * G);

    // Fold FC head into edge projections.
    fold_w_kernel<<<(H_DIM * FE_IN + 255) / 256, 256, 0, stream>>>(W_lp_e, b_lp_e, W_fc_lp, we_lp, be_lp);
    fold_w_kernel<<<(H_DIM * FE_IN + 255) / 256, 256, 0, stream>>>(W_pl_e, b_pl_e, W_fc_pl, we_pl, be_pl);

    // Node projections (fp32 WMMA).
    const size_t proj_lds = (size_t)(H_DIM * (F_IN + WPAD) + H_DIM + 16 * (F_IN + WPAD)) * sizeof(float);
    const int tilesL = (NL + 15) / 16;
    const int tilesP = (NP + 15) / 16;
    proj_gemm_kernel<<<imin(tilesL, 1024), 256, proj_lds, stream>>>(ligand_h, NL, W_lp_src, b_lp_src, h_lp_s, tilesL);
    proj_gemm_kernel<<<imin(tilesP, 1024), 256, proj_lds, stream>>>(pocket_h, NP, W_lp_dst, b_lp_dst, h_lp_d, tilesP);
    proj_gemm_kernel<<<imin(tilesP, 1024), 256, proj_lds, stream>>>(pocket_h, NP, W_pl_src, b_pl_src, h_pl_s, tilesP);
    proj_gemm_kernel<<<imin(tilesL, 1024), 256, proj_lds, stream>>>(ligand_h, NL, W_pl_dst, b_pl_dst, h_pl_d, tilesL);

    // Fused edge kernels (disjoint output halves; lp then pl).
    const size_t edge_lds = (size_t)(H_DIM * (FE_IN + EPAD) + H_DIM + G_MAX + 8 * 16 * (FE_IN + EPAD)) * sizeof(float);
    const int tilesLP = (ELP + 15) / 16;
    const int tilesPL = (EPL + 15) / 16;
    edge_fused_kernel<<<imin((tilesLP + 7) / 8, 2048), 256, edge_lds, stream>>>(
        e_lp, src_lp, dst_lp, gid_lp, we_lp, be_lp, h_lp_s, h_lp_d, b_fc_lp,
        (float*)d_out, ELP, G);
    edge_fused_kernel<<<imin((tilesPL + 7) / 8, 2048), 256, edge_lds, stream>>>(
        e_pl, src_pl, dst_pl, gid_pl, we_pl, be_pl, h_pl_s, h_pl_d, b_fc_pl,
        (float*)d_out + G, EPL, G);
}